// HNNLayer_85126251807356
// MI455X (gfx1250) — compile-verified
//
#include <hip/hip_runtime.h>

typedef __attribute__((ext_vector_type(2))) float v2f;
typedef __attribute__((ext_vector_type(8))) float v8f;

#define LDSPAD 132

// ---------------------------------------------------------------------------
// Generic 128-K, 128-N fp32 WMMA GEMM: out[M,128] = epilogue(X1@W1^T, X2@W2^T)
//   mode 0: out = relu(X1@W1^T)
//   mode 1: out = X1@W1^T + rs1[row]*(X2@W2^T + bias[col])
//   mode 2: out = rs1[row]*(X1@W1^T + rs2[row]*(X2@W2^T + bias[col]))
// W element [n][k] at W[n*wstride + k]. One block = 16 rows x 128 cols,
// 8 waves, each wave owns a 16x16 tile via V_WMMA_F32_16X16X4_F32 (32 steps).
// M must be a multiple of 16 (holds: 50000, 10000).
// ---------------------------------------------------------------------------
__global__ __launch_bounds__(256)
void gemm128_wmma(const float* __restrict__ X1, const float* __restrict__ X2,
                  const float* __restrict__ W1, const float* __restrict__ W2,
                  int wstride,
                  const float* __restrict__ bias,
                  const float* __restrict__ rs1, const float* __restrict__ rs2,
                  float* __restrict__ out, int mode)
{
    __shared__ float Xs1[16 * LDSPAD];
    __shared__ float Xs2[16 * LDSPAD];

    const int tid  = threadIdx.x;
    const int wave = tid >> 5;
    const int lane = tid & 31;
    const int lo   = lane & 15;
    const int hi   = lane >> 4;
    const int row0 = blockIdx.x << 4;
    const int col0 = wave << 4;

    // Stage 16x128 X tile(s) into LDS (coalesced).
#pragma unroll
    for (int i = 0; i < 8; ++i) {
        int idx = tid + (i << 8);
        int r = idx >> 7, c = idx & 127;
        Xs1[r * LDSPAD + c] = X1[(size_t)(row0 + r) * 128 + c];
    }
    if (mode != 0) {
#pragma unroll
        for (int i = 0; i < 8; ++i) {
            int idx = tid + (i << 8);
            int r = idx >> 7, c = idx & 127;
            Xs2[r * LDSPAD + c] = X2[(size_t)(row0 + r) * 128 + c];
        }
    }
    __syncthreads();

    v8f acc1 = {0.f,0.f,0.f,0.f,0.f,0.f,0.f,0.f};
    v8f acc2 = {0.f,0.f,0.f,0.f,0.f,0.f,0.f,0.f};

    // A frag (16x4 f32): lanes 0-15 -> K={k,k+1}, lanes 16-31 -> K={k+2,k+3}, M=lo.
    // B frag (4x16 f32): same K split, N=lo; B[k][n] = W[n][k].
    const float* w1row = W1 + (size_t)(col0 + lo) * wstride + 2 * hi;
    const float* xs1   = &Xs1[lo * LDSPAD + 2 * hi];
#pragma unroll
    for (int k = 0; k < 128; k += 4) {
        v2f a = *(const v2f*)(xs1 + k);
        v2f b = *(const v2f*)(w1row + k);
        acc1 = __builtin_amdgcn_wmma_f32_16x16x4_f32(false, a, false, b,
                                                     (short)0, acc1, false, false);
    }
    if (mode != 0) {
        const float* w2row = W2 + (size_t)(col0 + lo) * wstride + 2 * hi;
        const float* xs2   = &Xs2[lo * LDSPAD + 2 * hi];
#pragma unroll
        for (int k = 0; k < 128; k += 4) {
            v2f a = *(const v2f*)(xs2 + k);
            v2f b = *(const v2f*)(w2row + k);
            acc2 = __builtin_amdgcn_wmma_f32_16x16x4_f32(false, a, false, b,
                                                         (short)0, acc2, false, false);
        }
    }

    // D layout: VGPR v -> rows (row0+v) [lanes 0-15] / (row0+v+8) [lanes 16-31].
    const int colg = col0 + lo;
    const float bv = (mode != 0) ? bias[colg] : 0.f;
#pragma unroll
    for (int v = 0; v < 8; ++v) {
        int row = row0 + v + (hi << 3);
        float r;
        if (mode == 0)      { r = acc1[v] > 0.f ? acc1[v] : 0.f; }
        else if (mode == 1) { r = acc1[v] + rs1[row] * (acc2[v] + bv); }
        else                { r = rs1[row] * (acc1[v] + rs2[row] * (acc2[v] + bv)); }
        out[(size_t)row * 128 + colg] = r;
    }
}

// ---------------------------------------------------------------------------
// dst[widx[j], :] += (vals?vals[j]:1) * src[ridx[j], :]   (rows of 128 f32)
// One wave per nnz entry; 4 floats per lane; fp32 global atomics.
// ---------------------------------------------------------------------------
__global__ __launch_bounds__(256)
void scatter_rows(const float* __restrict__ src, const int* __restrict__ ridx,
                  const int* __restrict__ widx, const float* __restrict__ vals,
                  float* __restrict__ dst, int nnz)
{
    int j = blockIdx.x * 8 + (threadIdx.x >> 5);
    if (j >= nnz) return;
    int lane = threadIdx.x & 31;
    int r = ridx[j], w = widx[j];
    float s = vals ? vals[j] : 1.0f;
    const float* sp = src + (size_t)r * 128;
    float*       dp = dst + (size_t)w * 128;
#pragma unroll
    for (int i = 0; i < 4; ++i)
        atomicAdd(&dp[lane + 32 * i], s * sp[lane + 32 * i]);
}

// ---------------------------------------------------------------------------
// psi1 pre-aggregation:
//   U[dst]  += invDV[src] * vfeat[src]   (E x 128)
//   c[dst]  += invDV[src]
//   deg[dst]+= 1
// ---------------------------------------------------------------------------
__global__ __launch_bounds__(256)
void scatter_psi1(const float* __restrict__ vfeat, const float* __restrict__ invDV,
                  const int* __restrict__ src, const int* __restrict__ dst,
                  float* __restrict__ U, float* __restrict__ c,
                  float* __restrict__ deg, int nnz)
{
    int j = blockIdx.x * 8 + (threadIdx.x >> 5);
    if (j >= nnz) return;
    int lane = threadIdx.x & 31;
    int s = src[j], d = dst[j];
    float w = invDV[s];
    const float* vp = vfeat + (size_t)s * 128;
    float*       up = U + (size_t)d * 128;
#pragma unroll
    for (int i = 0; i < 4; ++i)
        atomicAdd(&up[lane + 32 * i], w * vp[lane + 32 * i]);
    if (lane == 0) {
        atomicAdd(&c[d], w);
        atomicAdd(&deg[d], 1.0f);
    }
}

__global__ void fill_zero_f32(float* __restrict__ p, long n)
{
    for (long i = (long)blockIdx.x * blockDim.x + threadIdx.x; i < n;
         i += (long)gridDim.x * blockDim.x)
        p[i] = 0.0f;
}

__global__ void copy_f32(float* __restrict__ d, const float* __restrict__ s, long n)
{
    for (long i = (long)blockIdx.x * blockDim.x + threadIdx.x; i < n;
         i += (long)gridDim.x * blockDim.x)
        d[i] = s[i];
}

static inline int gsz(long n, int per) { return (int)((n + per - 1) / per); }

extern "C" void kernel_launch(void* const* d_in, const int* in_sizes, int n_in,
                              void* d_out, int out_size, void* d_ws, size_t ws_size,
                              hipStream_t stream)
{
    const float* vfeat    = (const float*)d_in[0];
    const float* efeat    = (const float*)d_in[1];
    const float* invDV    = (const float*)d_in[2];
    const float* invDE    = (const float*)d_in[3];
    const int*   inc_src  = (const int*)d_in[4];
    const int*   inc_dst  = (const int*)d_in[5];
    const int*   emat_r   = (const int*)d_in[6];
    const int*   emat_c   = (const int*)d_in[7];
    const float* emat_v   = (const float*)d_in[8];
    const int*   vmat_r   = (const int*)d_in[9];
    const int*   vmat_c   = (const int*)d_in[10];
    const float* vmat_v   = (const float*)d_in[11];
    const float* Wv       = (const float*)d_in[12];
    const float* We       = (const float*)d_in[13];
    const float* psi1_W   = (const float*)d_in[14];
    const float* psi1_b   = (const float*)d_in[15];
    const float* psi2_W   = (const float*)d_in[16];
    const float* psi2_b   = (const float*)d_in[17];

    const int N    = in_sizes[2];   // 50000
    const int E    = in_sizes[3];   // 10000
    const int NNZ  = in_sizes[4];   // 400000
    const int NNZE = in_sizes[6];   // 80000
    const int NNZV = in_sizes[9];   // 400000

    const long E128 = (long)E * 128;
    const long N128 = (long)N * 128;

    float* ws   = (float*)d_ws;
    float* U    = ws;                 // E x 128   (later reused as S)
    float* Apre = ws + E128;          // E x 128   (later reused as B)
    float* Aef  = ws + 2 * E128;      // E x 128   (_efeat, later Eg2)
    float* cbuf = ws + 3 * E128;      // E
    float* degb = cbuf + E;           // E
    float* Vg   = degb + E;           // N x 128   (later reused as Vg2)
    float* Vm   = Vg + N128;          // N x 128

    float* vout = (float*)d_out;          // N x 128
    float* eout = vout + N128;            // E x 128

    dim3 blk(256);
    const int zg = 4096;

    // ---- psi1 pre-aggregation: U, c, deg ----
    fill_zero_f32<<<zg, blk, 0, stream>>>(U, E128);
    fill_zero_f32<<<gsz(2 * E, 256), blk, 0, stream>>>(cbuf, 2L * E);
    scatter_psi1<<<gsz(NNZ, 8), blk, 0, stream>>>(vfeat, invDV, inc_src, inc_dst,
                                                  U, cbuf, degb, NNZ);

    // ---- Apre = U@P1v^T + c * (efeat@P1e^T + b1)  (mode 1) ----
    gemm128_wmma<<<E / 16, blk, 0, stream>>>(U, efeat, psi1_W, psi1_W + 128, 256,
                                             psi1_b, cbuf, nullptr, Apre, 1);

    // ---- _efeat = efeat + emat @ Apre ----
    copy_f32<<<zg, blk, 0, stream>>>(Aef, efeat, E128);
    scatter_rows<<<gsz(NNZE, 8), blk, 0, stream>>>(Apre, emat_c, emat_r, emat_v,
                                                   Aef, NNZE);

    // ---- _vfeat = H @ _efeat ; vfeat_out = relu(_vfeat @ Wv^T) ----
    fill_zero_f32<<<zg, blk, 0, stream>>>(Vg, N128);
    scatter_rows<<<gsz(NNZ, 8), blk, 0, stream>>>(Aef, inc_dst, inc_src, nullptr,
                                                  Vg, NNZ);
    gemm128_wmma<<<N / 16, blk, 0, stream>>>(Vg, nullptr, Wv, nullptr, 128,
                                             nullptr, nullptr, nullptr, vout, 0);

    // ---- S = H^T @ vfeat_out ; B = invDE*(S@P2v^T + deg*(efeat@P2e^T + b2)) ----
    float* S = U;      // U dead
    float* B = Apre;   // Apre dead
    fill_zero_f32<<<zg, blk, 0, stream>>>(S, E128);
    scatter_rows<<<gsz(NNZ, 8), blk, 0, stream>>>(vout, inc_src, inc_dst, nullptr,
                                                  S, NNZ);
    gemm128_wmma<<<E / 16, blk, 0, stream>>>(S, efeat, psi2_W, psi2_W + 128, 256,
                                             psi2_b, invDE, degb, B, 2);

    // ---- _vfeat2 = vmat @ (H @ efeat) ----
    float* Vg2 = Vg;   // Vg dead
    fill_zero_f32<<<zg, blk, 0, stream>>>(Vg2, N128);
    scatter_rows<<<gsz(NNZ, 8), blk, 0, stream>>>(efeat, inc_dst, inc_src, nullptr,
                                                  Vg2, NNZ);
    fill_zero_f32<<<zg, blk, 0, stream>>>(Vm, N128);
    scatter_rows<<<gsz(NNZV, 8), blk, 0, stream>>>(Vg2, vmat_c, vmat_r, vmat_v,
                                                   Vm, NNZV);

    // ---- _efeat2 = B + H^T @ _vfeat2 ; efeat_out = relu(_efeat2 @ We^T) ----
    float* Eg2 = Aef;  // Aef dead
    copy_f32<<<zg, blk, 0, stream>>>(Eg2, B, E128);
    scatter_rows<<<gsz(NNZ, 8), blk, 0, stream>>>(Vm, inc_src, inc_dst, nullptr,
                                                  Eg2, NNZ);
    gemm128_wmma<<<E / 16, blk, 0, stream>>>(Eg2, nullptr, We, nullptr, 128,
                                             nullptr, nullptr, nullptr, eout, 0);
}